// PointsLoss_31748398252573
// MI455X (gfx1250) — compile-verified
//
#include <hip/hip_runtime.h>
#include <hip/hip_bf16.h>
#include <math.h>

typedef float v2f __attribute__((ext_vector_type(2)));
typedef float v8f __attribute__((ext_vector_type(8)));

#define NPTS 65536   // 256*256
#define NCH  128

// ---------------------------------------------------------------------------
// Stage 1: channel reduction via V_WMMA_F32_16X16X4_F32.
// A = all-ones 16x4  =>  D[m,n] = sum_k B[k,n]; each WMMA sums 4 channels
// for 16 points. Two chains per wave -> 32 points/wave, all lanes store.
// 8192 waves total (2 tensors x 2 batches x 2048 tiles).
// Streaming inputs are read once (134 MB in total, no reuse) -> non-temporal loads.
// ---------------------------------------------------------------------------
__global__ __launch_bounds__(256)
void channel_sum_wmma(const float* __restrict__ added,   // [2,128,65536]
                      const float* __restrict__ orig,    // [2,129,65536]
                      float* __restrict__ pred,          // [2,65536]
                      float* __restrict__ osum)          // [2,65536]
{
    const int tid    = blockIdx.x * blockDim.x + threadIdx.x;
    const int wave   = tid >> 5;            // 0..8191
    const int lane   = tid & 31;
    const int tensor = wave >> 12;          // 0 = added, 1 = orig
    const int rem    = wave & 4095;
    const int b      = rem >> 11;           // batch
    const int tile   = rem & 2047;
    const int p0     = tile << 5;           // 32 points per wave

    const int half = lane >> 4;             // which half-wave
    const int ln   = lane & 15;

    const float* src = (tensor == 0)
        ? added + (size_t)b * NCH * NPTS
        : orig  + ((size_t)b * (NCH + 1) + 1) * NPTS;   // skip channel 0

    const v2f a = {1.0f, 1.0f};             // all-ones A matrix
    v8f acc0 = {};                          // chain0: points p0..p0+15
    v8f acc1 = {};                          // chain1: points p0+16..p0+31

    const float* s0 = src + p0 + ln;
    const float* s1 = s0 + 16;

#pragma unroll 4
    for (int c0 = 0; c0 < NCH; c0 += 4) {
        const size_t coff = (size_t)(c0 + 2 * half) * NPTS;
        v2f b0, b1;
        b0.x = __builtin_nontemporal_load(s0 + coff);
        b0.y = __builtin_nontemporal_load(s0 + coff + NPTS);
        b1.x = __builtin_nontemporal_load(s1 + coff);
        b1.y = __builtin_nontemporal_load(s1 + coff + NPTS);
        acc0 = __builtin_amdgcn_wmma_f32_16x16x4_f32(
            false, a, false, b0, (short)0, acc0, false, false);
        acc1 = __builtin_amdgcn_wmma_f32_16x16x4_f32(
            false, a, false, b1, (short)0, acc1, false, false);
    }

    // D row m lives in: lanes 0-15 VGPR0 -> M=0,N=lane ; lanes 16-31 VGPR0 -> M=8,N=lane-16.
    // Rows are identical (A all-ones), so lane<16 reads chain0, lane>=16 reads chain1.
    float* dst = ((tensor == 0) ? pred : osum) + (size_t)b * NPTS + p0;
    dst[lane] = (half == 0) ? acc0[0] : acc1[0];
}

// ---------------------------------------------------------------------------
// Stage 2: point-in-rotated-box + scatter-max onto 256x256 bool grids
// (stored as 2048 u32 bitmask words per grid, set with atomicOr).
// ---------------------------------------------------------------------------
__global__ __launch_bounds__(256)
void mask_scatter(const float* __restrict__ pred,
                  const float* __restrict__ osum,
                  const float* __restrict__ boxes,     // [2,20,7]
                  unsigned* __restrict__ gridP,        // [2,2048]
                  unsigned* __restrict__ gridO)        // [2,2048]
{
    __shared__ float bx[20][8];   // cx, cy, hx, hy, cos(-h), sin(-h), cz, hz
    const int b = blockIdx.x >> 8;
    const int p = ((blockIdx.x & 255) << 8) | threadIdx.x;

    if (threadIdx.x < 20) {
        const float* B = boxes + ((size_t)b * 20 + threadIdx.x) * 7;
        const float h = B[6];
        bx[threadIdx.x][0] = B[0];
        bx[threadIdx.x][1] = B[1];
        bx[threadIdx.x][2] = 0.5f * B[3];
        bx[threadIdx.x][3] = 0.5f * B[4];
        bx[threadIdx.x][4] = cosf(-h);
        bx[threadIdx.x][5] = sinf(-h);
        bx[threadIdx.x][6] = B[2];
        bx[threadIdx.x][7] = 0.5f * B[5];
    }
    __syncthreads();

    const int i = p >> 8, j = p & 255;
    const float x = i * 0.8f, y = j * 0.8f, z = 0.8f;

    bool inside = false;
    for (int t = 0; t < 20; ++t) {
        const float dx = x - bx[t][0];
        const float dy = y - bx[t][1];
        const float dz = z - bx[t][6];
        const float c  = bx[t][4], s = bx[t][5];
        const float lx = dx * c - dy * s;
        const float ly = dx * s + dy * c;
        inside |= (fabsf(lx) <= bx[t][2]) & (fabsf(ly) <= bx[t][3]) &
                  (fabsf(dz) <= bx[t][7]);
    }

    const size_t bp = (size_t)b * NPTS + p;
    const bool mp = inside && (pred[bp] != 0.0f);
    const bool mo = inside && (osum[bp] != 0.0f);

    if (mp | mo) {
        const int gi  = min(i + 128, 255);
        const int gj  = min(j + 128, 255);
        const int idx = (gi << 8) | gj;
        const unsigned bit = 1u << (idx & 31);
        if (mp) atomicOr(&gridP[(b << 11) + (idx >> 5)], bit);
        if (mo) atomicOr(&gridO[(b << 11) + (idx >> 5)], bit);
    }
}

// ---------------------------------------------------------------------------
// Stage 3: popcount reduction -> IoU per batch -> mean. One block.
// ---------------------------------------------------------------------------
__global__ __launch_bounds__(256)
void iou_reduce(const unsigned* __restrict__ gridP,
                const unsigned* __restrict__ gridO,
                float* __restrict__ out)
{
    __shared__ int sI[256], sU[256];
    float iouSum = 0.0f;

    for (int b = 0; b < 2; ++b) {
        int inter = 0, uni = 0;
        for (int w = threadIdx.x; w < 2048; w += 256) {
            const unsigned pw = gridP[(b << 11) + w];
            const unsigned ow = gridO[(b << 11) + w];
            inter += __popc(pw & ow);
            uni   += __popc(pw | ow);
        }
        sI[threadIdx.x] = inter;
        sU[threadIdx.x] = uni;
        __syncthreads();
        for (int off = 128; off > 0; off >>= 1) {
            if (threadIdx.x < off) {
                sI[threadIdx.x] += sI[threadIdx.x + off];
                sU[threadIdx.x] += sU[threadIdx.x + off];
            }
            __syncthreads();
        }
        if (threadIdx.x == 0) {
            const float fi = (float)sI[0], fu = (float)sU[0];
            iouSum += (fu > 0.0f) ? fi / fmaxf(fu, 1.0f) : 0.0f;
        }
        __syncthreads();
    }
    if (threadIdx.x == 0) out[0] = 0.5f * iouSum;
}

// ---------------------------------------------------------------------------
extern "C" void kernel_launch(void* const* d_in, const int* in_sizes, int n_in,
                              void* d_out, int out_size, void* d_ws, size_t ws_size,
                              hipStream_t stream)
{
    const float* added = (const float*)d_in[0];   // [2,128,256,256]
    const float* orig  = (const float*)d_in[1];   // [2,129,256,256]
    const float* boxes = (const float*)d_in[2];   // [2,20,7]
    float* out = (float*)d_out;

    float*    pred  = (float*)d_ws;                 // 2*65536 f32
    float*    osum  = pred + 2 * NPTS;              // 2*65536 f32
    unsigned* gridP = (unsigned*)(osum + 2 * NPTS); // 2*2048 u32
    unsigned* gridO = gridP + 2 * 2048;             // 2*2048 u32

    hipMemsetAsync(gridP, 0, 2 * 2 * 2048 * sizeof(unsigned), stream);

    channel_sum_wmma<<<1024, 256, 0, stream>>>(added, orig, pred, osum);
    mask_scatter<<<512, 256, 0, stream>>>(pred, osum, boxes, gridP, gridO);
    iou_reduce<<<1, 256, 0, stream>>>(gridP, gridO, out);
}